// EquiNN_1322849927394
// MI455X (gfx1250) — compile-verified
//
#include <hip/hip_runtime.h>

typedef __attribute__((ext_vector_type(2))) float v2f;
typedef __attribute__((ext_vector_type(4))) float v4f;
typedef __attribute__((ext_vector_type(8))) float v8f;

// One block per row of X[4096][8192]. 256 threads (8 wave32) per block.
// Each thread: 8 x float4 = 32 elements held in VGPRs (single HBM read).
// Row sum: per-thread partial -> v_wmma_f32_16x16x4_f32 cross-lane fold
// -> ds_swizzle SWAPX16 -> per-wave LDS slot -> one barrier -> broadcast.
__global__ __launch_bounds__(256) void equinn_row_kernel(
    const float* __restrict__ X,
    const float* __restrict__ lp,
    const float* __restrict__ gp,
    const float* __restrict__ bp,
    float* __restrict__ out)
{
    constexpr int COLS   = 8192;
    constexpr int VEC    = COLS / 4;      // 2048 float4 per row
    constexpr int NTHR   = 256;
    constexpr int PERTHR = VEC / NTHR;    // 8 float4 per thread

    const int tid = threadIdx.x;
    const size_t rowbase = (size_t)blockIdx.x * COLS;
    const v4f* __restrict__ Xr = (const v4f*)(X + rowbase);
    v4f*       __restrict__ Or = (v4f*)(out + rowbase);

    // ---- streaming load: X row lives in registers (read HBM once) ----
    v4f r[PERTHR];
    float p = 0.0f;
#pragma unroll
    for (int v = 0; v < PERTHR; ++v) {
        r[v] = __builtin_nontemporal_load(&Xr[tid + NTHR * v]);
        p += r[v].x + r[v].y + r[v].z + r[v].w;
    }

    // ---- wave32 reduction via WMMA f32 16x16x4 ----
    // A: lane L (0-15) -> A[L][0] = p_L ; lane L (16-31) -> A[L-16][2] = p_L ;
    // K-slots 1,3 zeroed. B = ones(4x16) => D[m][n] = p_m + p_{m+16} (all n).
    v2f a; a.x = p;    a.y = 0.0f;
    v2f bm; bm.x = 1.0f; bm.y = 1.0f;
    v8f c = {};
    c = __builtin_amdgcn_wmma_f32_16x16x4_f32(
        /*neg_a=*/false, a, /*neg_b=*/false, bm,
        /*c_mod=*/(short)0, c, /*reuse_a=*/false, /*reuse_b=*/false);

    // Lane n (<16): sum of c[0..7] = p_0..p_7 + p_16..p_23 ;
    // lane n+16:    sum of c[0..7] = p_8..p_15 + p_24..p_31.
    // One xor-16 exchange (ds_swizzle SWAPX16: {xor=0x10, and=0x1f} = 0x401F)
    // completes the 32-lane sum in every lane.
    float s = ((c[0] + c[1]) + (c[2] + c[3])) + ((c[4] + c[5]) + (c[6] + c[7]));
    {
        int si = __builtin_bit_cast(int, s);
        int sw = __builtin_amdgcn_ds_swizzle(si, 0x401F);
        s += __builtin_bit_cast(float, sw);
    }

    // ---- cross-wave combine: per-wave slot, one barrier, b128 reload ----
    __shared__ v4f wsum4[2];                  // 8 wave sums
    const int lane = tid & 31;
    const int wave = tid >> 5;
    if (lane == 0) ((float*)wsum4)[wave] = s; // ds_store_b32
    __syncthreads();
    const v4f w0 = wsum4[0];                  // ds_load_b128
    const v4f w1 = wsum4[1];                  // ds_load_b128
    const float rowsum = ((w0.x + w0.y) + (w0.z + w0.w))
                       + ((w1.x + w1.y) + (w1.z + w1.w));

    // ---- fused scale/add from registers, NT streaming store ----
    const float l = lp[0], g = gp[0], b = bp[0];
    const float add = g * rowsum + b;
#pragma unroll
    for (int v = 0; v < PERTHR; ++v) {
        v4f o;
        o.x = fmaf(l, r[v].x, add);
        o.y = fmaf(l, r[v].y, add);
        o.z = fmaf(l, r[v].z, add);
        o.w = fmaf(l, r[v].w, add);
        __builtin_nontemporal_store(o, &Or[tid + NTHR * v]);
    }
}

extern "C" void kernel_launch(void* const* d_in, const int* in_sizes, int n_in,
                              void* d_out, int out_size, void* d_ws, size_t ws_size,
                              hipStream_t stream) {
    (void)in_sizes; (void)n_in; (void)d_ws; (void)ws_size; (void)out_size;
    const float* X = (const float*)d_in[0];
    const float* l = (const float*)d_in[1];
    const float* g = (const float*)d_in[2];
    const float* b = (const float*)d_in[3];
    float* out = (float*)d_out;

    // 4096 rows, one block per row, 256 threads (8 wave32) per block.
    equinn_row_kernel<<<4096, 256, 0, stream>>>(X, l, g, b, out);
}